// Transformer_30030411333994
// MI455X (gfx1250) — compile-verified
//
#include <hip/hip_runtime.h>

typedef __attribute__((ext_vector_type(2))) float v2f;
typedef __attribute__((ext_vector_type(8))) float v8f;

#define B_DIM 256
#define L_DIM 512
#define H_DIM 256
#define K_DIM 512

// One block per (k-tile of 128, b). 8 waves/block, each wave computes a
// 16-k tile of preds via V_WMMA_F32_16X16X4_F32 accumulation over H.
__global__ __launch_bounds__(256) void preds_gather_wmma_kernel(
    const int* __restrict__ entities,       // [B, L]
    const int* __restrict__ actions,        // [A]
    const int* __restrict__ action_keys,    // [B, K]
    const float* __restrict__ encoded,      // [B, L, H]
    const float* __restrict__ rel_emb,      // [REL_VOCAB, H]
    float* __restrict__ out)                // [B, K]
{
    __shared__ int cnt[256];
    __shared__ __align__(16) float xbuf[H_DIM];

    const int t = threadIdx.x;
    const int b = blockIdx.y;

    // ---- last non-pad position: idx = popcount(entities[b,:] != 0) - 1 ----
    const int* erow = entities + (size_t)b * L_DIM;
    cnt[t] = (erow[t] != 0 ? 1 : 0) + (erow[t + 256] != 0 ? 1 : 0);
    __syncthreads();
    for (int s = 128; s > 0; s >>= 1) {
        if (t < s) cnt[t] += cnt[t + s];
        __syncthreads();
    }
    int idx = cnt[0] - 1;
    if (idx < 0) idx = 0;

    // ---- stage enc_last[b,:] (256 f32 = 1KB) in LDS ----
    xbuf[t] = encoded[((size_t)b * L_DIM + idx) * H_DIM + t];
    __syncthreads();

    const int wave = t >> 5;
    const int lane = t & 31;
    const int grp  = lane >> 4;   // half-wave group: K-rows {0,1} vs {2,3}
    const int m    = lane & 15;   // A-matrix row (k within tile)
    const int k0   = blockIdx.x * 128 + wave * 16;

    // ---- double-indirect gather of the 16 embedding rows for this tile ----
    // lanes l and l+16 share a row; each covers a different K-slice pair.
    const int key = action_keys[(size_t)b * K_DIM + k0 + m];
    const int act = actions[key];
    const float* arow = rel_emb + (size_t)act * H_DIM + 2 * grp;
    const float* xrow = xbuf + 2 * grp;

    // ---- 64 x V_WMMA_F32_16X16X4_F32: D = A(16x4) * B(4x16, bcast) + C ----
    v8f acc = {};
#pragma unroll 8
    for (int h = 0; h < H_DIM; h += 4) {
        v2f a  = *(const v2f*)(arow + h);   // global_load_b64 (L2-resident gather)
        v2f bx = *(const v2f*)(xrow + h);   // ds_load_b64 broadcast
        acc = __builtin_amdgcn_wmma_f32_16x16x4_f32(
            /*neg_a=*/false, a, /*neg_b=*/false, bx,
            /*c_mod=*/(short)0, acc, /*reuse_a=*/false, /*reuse_b=*/false);
    }

    // ---- column N=0 holds the preds: lane 0 -> k0+0..7, lane 16 -> k0+8..15
    if (m == 0) {
        float* o = out + (size_t)b * K_DIM + k0 + grp * 8;
        *(float4*)(o)     = make_float4(acc[0], acc[1], acc[2], acc[3]);
        *(float4*)(o + 4) = make_float4(acc[4], acc[5], acc[6], acc[7]);
    }
}

extern "C" void kernel_launch(void* const* d_in, const int* in_sizes, int n_in,
                              void* d_out, int out_size, void* d_ws, size_t ws_size,
                              hipStream_t stream) {
    (void)in_sizes; (void)n_in; (void)out_size; (void)d_ws; (void)ws_size;

    const int*   entities    = (const int*)d_in[0];
    /* d_in[1] = relations: unused by the reference computation */
    const int*   actions     = (const int*)d_in[2];
    const int*   action_keys = (const int*)d_in[3];
    const float* encoded     = (const float*)d_in[4];
    const float* rel_emb     = (const float*)d_in[5];
    float*       out         = (float*)d_out;

    dim3 grid(K_DIM / 128, B_DIM);   // (4, 256)
    preds_gather_wmma_kernel<<<grid, 256, 0, stream>>>(
        entities, actions, action_keys, encoded, rel_emb, out);
}